// UncompressTransformLayer_85366769975790
// MI455X (gfx1250) — compile-verified
//
#include <hip/hip_runtime.h>
#include <cmath>

// Pure scatter: compressed[m] -> strict upper triangle of out[n][n], zeros elsewhere.
// n = 8192, m = n(n-1)/2. ~403 MB of HBM traffic, 0 FLOPs -> ~17us floor @ 23.3 TB/s.
// No matmul => WMMA not applicable. CDNA5 features used: async global<->LDS DMA
// (ASYNCcnt + s_wait_asynccnt) for the bulk copy, b128 stores, non-temporal hints.

typedef float v4f  __attribute__((ext_vector_type(4)));
typedef int   vi4  __attribute__((vector_size(4 * sizeof(int))));  // matches builtin param

typedef __attribute__((address_space(1))) int  g_i32;
typedef __attribute__((address_space(3))) int  l_i32;
typedef __attribute__((address_space(1))) vi4  g_vi4;
typedef __attribute__((address_space(3))) vi4  l_vi4;

#define COLS_PER_BLOCK 1024  // 256 threads x 4 floats

#if defined(__has_builtin)
#if __has_builtin(__builtin_amdgcn_global_load_async_to_lds_b32) && \
    __has_builtin(__builtin_amdgcn_global_store_async_from_lds_b128) && \
    __has_builtin(__builtin_amdgcn_s_wait_asynccnt)
#define HAVE_ASYNC_LDS 1
#endif
#endif
#ifndef HAVE_ASYNC_LDS
#define HAVE_ASYNC_LDS 0
#endif

__global__ void __launch_bounds__(256)
triu_scatter_kernel(const float* __restrict__ comp, float* __restrict__ out, int n) {
    const int i     = blockIdx.y;                    // output row
    const int cbase = blockIdx.x * COLS_PER_BLOCK;   // first column of this block
    const int tid   = threadIdx.x;
    const int c0    = cbase + tid * 4;               // this thread's 4 columns

    // Start of row i's data in the compressed array: S(i) = i*(n-1) - i*(i-1)/2
    const long long Si = (long long)i * (n - 1) - ((long long)i * (i - 1)) / 2;
    float* __restrict__ orow = out + (long long)i * n;

    const bool all_zero = (cbase + COLS_PER_BLOCK - 1) <= i;  // every col <= i
    const bool all_copy = (cbase >= i + 1);                   // every col  > i

    if (all_zero) {
        // Diagonal-and-below: stream zeros, non-temporal b128 stores.
        v4f z = (v4f)0.0f;
        __builtin_nontemporal_store(z, (v4f*)(orow + c0));
        return;
    }

    if (all_copy) {
        // Contiguous copy: compressed[Si + (cbase - i - 1) ...] -> orow[cbase ...]
        const float* __restrict__ src = comp + (Si + (long long)cbase - (i + 1));
#if HAVE_ASYNC_LDS
        // CDNA5 async DMA path: global -> LDS (b32, src only dword-aligned per row),
        // wait ASYNCcnt, then LDS -> global b128 (dst is 16B-aligned).
        // Each wave reads back only LDS it wrote itself -> no workgroup barrier.
        __shared__ float lbuf[COLS_PER_BLOCK];
        g_i32* gsrc = (g_i32*)(src + tid * 4);          // AS1, const cast away (load-only)
        l_i32* lds  = (l_i32*)(&lbuf[tid * 4]);         // AS3
        // INST_OFFSET applies to both LDS and global addresses (ISA 08 §4.4).
        __builtin_amdgcn_global_load_async_to_lds_b32(gsrc, lds, 0, 0);
        __builtin_amdgcn_global_load_async_to_lds_b32(gsrc, lds, 4, 0);
        __builtin_amdgcn_global_load_async_to_lds_b32(gsrc, lds, 8, 0);
        __builtin_amdgcn_global_load_async_to_lds_b32(gsrc, lds, 12, 0);
        __builtin_amdgcn_s_wait_asynccnt(0);
        g_vi4* gdst = (g_vi4*)(orow + c0);              // 16B-aligned global dst
        l_vi4* ldsv = (l_vi4*)(&lbuf[tid * 4]);         // 16B-aligned LDS src
        __builtin_amdgcn_global_store_async_from_lds_b128(gdst, ldsv, 0, 0);
#else
        v4f v;
        v.x = __builtin_nontemporal_load(src + tid * 4 + 0);
        v.y = __builtin_nontemporal_load(src + tid * 4 + 1);
        v.z = __builtin_nontemporal_load(src + tid * 4 + 2);
        v.w = __builtin_nontemporal_load(src + tid * 4 + 3);
        __builtin_nontemporal_store(v, (v4f*)(orow + c0));
#endif
        return;
    }

    // Mixed block (exactly one per row): straddles the diagonal, do it scalar.
    #pragma unroll
    for (int k = 0; k < 4; ++k) {
        const int c = c0 + k;
        float v = 0.0f;
        if (c > i) v = comp[Si + (long long)(c - i - 1)];
        orow[c] = v;
    }
}

extern "C" void kernel_launch(void* const* d_in, const int* in_sizes, int n_in,
                              void* d_out, int out_size, void* d_ws, size_t ws_size,
                              hipStream_t stream) {
    const float* comp = (const float*)d_in[0];
    float* out        = (float*)d_out;

    const long long m = (long long)in_sizes[0];
    const int n = (int)llround(sqrt(2.0 * (double)m)) + 1;  // 8192

    dim3 block(256);
    dim3 grid((unsigned)((n + COLS_PER_BLOCK - 1) / COLS_PER_BLOCK), (unsigned)n);
    triu_scatter_kernel<<<grid, block, 0, stream>>>(comp, out, n);
}